// MoDGPT_34428457844856
// MI455X (gfx1250) — compile-verified
//
#include <hip/hip_runtime.h>
#include <hip/hip_bf16.h>
#include <stdint.h>

// ---- CDNA5 vector types -----------------------------------------------------
typedef __attribute__((ext_vector_type(2)))  float        v2f;
typedef __attribute__((ext_vector_type(4)))  float        v4f;
typedef __attribute__((ext_vector_type(8)))  float        v8f;
typedef __attribute__((ext_vector_type(4)))  unsigned int u32x4;
typedef __attribute__((ext_vector_type(8)))  int          i32x8;
typedef __attribute__((ext_vector_type(4)))  int          i32x4;

// Tile geometry: 128 threads = 4 wave32; each wave owns 16 tokens -> 64 tokens/WG.
#define DIM            1024
#define TOKENS_PER_WG  64
#define THREADS        128
// TDM pads 2 dwords after every 256 dwords stored -> LDS row stride 1032 dwords.
#define ROWSTRIDE      1032
#define XTILE_DWORDS   (TOKENS_PER_WG * ROWSTRIDE)   // 66048 dwords = 258 KB
#define LDS_DWORDS     (DIM + XTILE_DWORDS)          // + 4 KB for halt_w

// element (t,k) of the padded tile, in dwords
__device__ __forceinline__ int pad_idx(int t, int k) {
  return t * ROWSTRIDE + k + ((k >> 8) << 1);
}

__global__ void init_ponder(float* p) { *p = 0.0f; }

__global__ void __launch_bounds__(THREADS)
act_fused_kernel(const float* __restrict__ x,
                 const float* __restrict__ halt_w,
                 const float* __restrict__ halt_b,
                 float* __restrict__ out,
                 float* __restrict__ ponder,
                 float inv_tokens)   // 0.01 / total_tokens
{
  extern __shared__ float lds[];
  float* halt_lds = lds;            // DIM floats
  float* x_lds    = lds + DIM;      // padded 64x1024 tile

  const int tile = blockIdx.x;
  const int tid  = threadIdx.x;
  const int wave = tid >> 5;
  const int lane = tid & 31;
  const int tok  = lane & 15;            // token within wave group (also B-col)
  const int khalf = (lane >> 4) << 1;    // K sub-offset: lanes 0-15 -> 0, 16-31 -> 2
  const int t0   = wave * 16;            // first token of this wave inside tile

  // -------- Stage the x tile into LDS with the Tensor Data Mover -------------
  if (wave == 0) {
    unsigned long long ga =
        (unsigned long long)(uintptr_t)(x + (size_t)tile * TOKENS_PER_WG * DIM);
    unsigned lds_off = (unsigned)(uintptr_t)(void*)x_lds;  // LDS byte offset (low 32 bits)

    u32x4 g0;
    g0[0] = 1u;                                          // count=1, user descriptor
    g0[1] = lds_off;                                     // lds_addr
    g0[2] = (unsigned)(ga & 0xffffffffu);                // global_addr[31:0]
    g0[3] = (unsigned)((ga >> 32) & 0x1ffffffu)          // global_addr[56:32]
          | (2u << 30);                                  // type=2 ("image")
    i32x8 g1;
    g1[0] = (int)((2u << 16)      // data_size = 4 bytes
                | (1u << 20)      // pad_enable
                | (7u << 22)      // pad_interval code 7 = every 256 dwords
                | (1u << 25));    // pad_amount  code 1 = 2 dwords
    g1[1] = (int)(1024u << 16);   // tensor_dim0[15:0] = 1024
    g1[2] = (int)(64u   << 16);   // tensor_dim0 hi = 0 | tensor_dim1 = 64
    g1[3] = (int)(1024u << 16);   // tensor_dim1 hi = 0 | tile_dim0 = 1024
    g1[4] = 64;                   // tile_dim1 = 64, tile_dim2 = 0
    g1[5] = 1024;                 // tensor_dim0_stride lo
    g1[6] = 0;                    // stride0 hi | tensor_dim1_stride lo (2D: unused)
    g1[7] = 0;
    i32x4 g2 = {0, 0, 0, 0};      // 2D tensor: groups 2/3 unused
    i32x4 g3 = {0, 0, 0, 0};
    i32x8 g4 = {0, 0, 0, 0, 0, 0, 0, 0};  // extra descriptor group (clang-23 6-arg form)
    __builtin_amdgcn_tensor_load_to_lds(g0, g1, g2, g3, g4, 0);
  }

  // halt_w -> LDS (overlaps with the TDM transfer; disjoint LDS range)
  for (int i = tid; i < DIM; i += THREADS) halt_lds[i] = halt_w[i];

  if (wave == 0) __builtin_amdgcn_s_wait_tensorcnt(0);
  __syncthreads();

  // -------- Halt matvec via V_WMMA_F32_16X16X4_F32 ---------------------------
  // A (16x4): row 0 = halt_w[k0..k0+3]  (lane0 holds K0,K1; lane16 holds K2,K3)
  // B (4x16): column n = token (t0+n)   (lane n: K0,K1; lane 16+n: K2,K3)
  // D[0][n] accumulates dot(token n, halt_w) -> acc[0] on lanes 0..15.
  v8f acc = {};
  const v2f zero2 = {0.0f, 0.0f};
  for (int k0 = 0; k0 < DIM; k0 += 4) {
    const int kh = k0 + khalf;
    v2f hv = *(const v2f*)(halt_lds + kh);            // ds_load_b64 (broadcast)
    v2f a  = (tok == 0) ? hv : zero2;                 // only A-row 0 is live
    v2f b  = *(const v2f*)(x_lds + pad_idx(t0 + tok, kh));
    acc = __builtin_amdgcn_wmma_f32_16x16x4_f32(
        /*neg_a=*/false, a, /*neg_b=*/false, b,
        /*c_mod=*/(short)0, acc, /*reuse_a=*/false, /*reuse_b=*/false);
  }

  // -------- ACT recurrence collapses to a per-token scalar -------------------
  const float dot = acc[0];                  // valid on lanes 0..15
  const float hb  = halt_b[0];
  const float h   = 1.0f / (1.0f + __expf(-(dot + hb)));
  float accp = 0.0f, rem = 1.0f, coef = 0.0f, nupd = 0.0f;
#pragma unroll
  for (int s = 0; s < 3; ++s) {              // states are identical -> h constant
    float still   = (accp < 0.99f) ? 1.0f : 0.0f;
    float nacc    = accp + h * still;
    float use_rem = ((nacc > 0.99f) ? 1.0f : 0.0f) * still;
    float use_h   = (1.0f - use_rem) * still;
    float p       = use_h * h + use_rem * rem;
    coef += p;
    accp += p * still;
    rem  -= p * still;
    nupd += still;
  }

  // ponder loss: wave-reduce n_upd, one float atomic per wave
  float contrib = (lane < 16) ? nupd : 0.0f;
#pragma unroll
  for (int off = 16; off > 0; off >>= 1) contrib += __shfl_xor(contrib, off, 32);
  if (lane == 0) atomicAdd(ponder, contrib * inv_tokens);

  // -------- Scale pass: out[row,:] = coef(row) * x[row,:] --------------------
  const size_t tile_base = (size_t)tile * TOKENS_PER_WG * DIM;
  for (int r = 0; r < 16; ++r) {
    const float cr  = __shfl(coef, r, 32);
    const int   row = t0 + r;
    float* orow = out + tile_base + (size_t)row * DIM;
    for (int c = lane * 4; c < DIM; c += 128) {
      const int cd = pad_idx(row, c);
      v2f p0 = *(const v2f*)(lds + DIM + cd);       // ds_load_b64 x2 (pad keeps
      v2f p1 = *(const v2f*)(lds + DIM + cd + 2);   //  8B alignment & bank spread)
      v4f v  = {p0.x * cr, p0.y * cr, p1.x * cr, p1.y * cr};
      __builtin_nontemporal_store(v, (v4f*)(orow + c));  // NT: keep x hot in L2
    }
  }
}

extern "C" void kernel_launch(void* const* d_in, const int* in_sizes, int n_in,
                              void* d_out, int out_size, void* d_ws, size_t ws_size,
                              hipStream_t stream) {
  (void)n_in; (void)d_ws; (void)ws_size;
  const float* x        = (const float*)d_in[0];
  const float* router_w = (const float*)d_in[1];  // dead code after algebraic
  (void)router_w;                                 // simplification (identity block)
  const float* halt_w   = (const float*)d_in[2];
  const float* halt_b   = (const float*)d_in[3];
  float* out    = (float*)d_out;
  float* ponder = out + (out_size - 1);

  const int tokens = in_sizes[0] / DIM;           // 32768
  const int blocks = tokens / TOKENS_PER_WG;      // 512
  const float inv_tokens = 0.01f / (float)tokens;

  init_ponder<<<1, 1, 0, stream>>>(ponder);
  act_fused_kernel<<<blocks, THREADS, LDS_DWORDS * sizeof(float), stream>>>(
      x, halt_w, halt_b, out, ponder, inv_tokens);
}